// slicing_21861383536758
// MI455X (gfx1250) — compile-verified
//
#include <hip/hip_runtime.h>

typedef __attribute__((ext_vector_type(2))) float v2f;
typedef __attribute__((ext_vector_type(8))) float v8f;

static constexpr int B_ = 16;
static constexpr int C_ = 64;
static constexpr int H_ = 512;
static constexpr int W_ = 60;
static constexpr long HW_ = (long)H_ * W_;          // 30720
static constexpr int KTOT = 192;                    // 3 * 64, K = kh*64 + ci

// ---------------------------------------------------------------------------
// Stage weights into LDS, layout: wlds[(K/2)*128 + co*2 + (K&1)] = w[co,ci,kh,1]
// so an A-fragment (lane: M=lane%16, K=k0 + v + 2*(lane/16)) is ONE aligned
// v2f (ds_load_b64) at wlds + (k0/2)*128 + t*32 + (hi*128 + n*2).
// ---------------------------------------------------------------------------
__device__ __forceinline__ void load_weights_lds(const float* __restrict__ wt,
                                                 float* wlds) {
  for (int idx = threadIdx.x; idx < KTOT * 64; idx += 256) {
    const int K  = idx >> 6;          // 0..191
    const int co = idx & 63;
    const int kh = K >> 6;            // K = kh*64 + ci
    const int ci = K & 63;
    // w shape [C_out=64][C_in=64][3][3]; only kw==1 column matters
    wlds[(K >> 1) * 128 + co * 2 + (K & 1)] = wt[co * 576 + ci * 9 + kh * 3 + 1];
  }
  __syncthreads();
}

// ---------------------------------------------------------------------------
// Core: 64x16 fp32 GEMM tile via V_WMMA_F32_16X16X4_F32.
//   A = W_eff [co=64 x K=192] from LDS
//   B = src patch [K=192 x n=16], column n -> (b, h, i0+n), K=(kh,ci)
// acc[t] holds co-tile t (co = t*16 + v + 8*hi, n = lane%16).
// h-padding guard is wave-uniform -> whole kh block (and its WMMAs) skipped.
// ---------------------------------------------------------------------------
__device__ __forceinline__ void conv_tile(const float* __restrict__ src,
                                          const float* wlds,
                                          int b, int h, int i0,
                                          int n, int hi, v8f acc[4]) {
  const long laneB0 = (long)(2 * hi) * HW_ + n;   // v=0 element: ci += 2*hi
  const long laneB1 = laneB0 + HW_;               // v=1 element: ci += 1 more
  const int  aLane  = hi * 128 + n * 2;

#pragma unroll
  for (int kh = 0; kh < 3; ++kh) {
    const int hr = h + kh - 1;
    if (hr < 0 || hr >= H_) continue;             // uniform per wave: skip padded row
    const float* srow = src + ((long)b * C_ * H_ + hr) * W_ + i0;
#pragma unroll
    for (int kc = 0; kc < 16; ++kc) {
      const long cbase = (long)(kc * 4) * HW_;    // ci base for this K step
      v2f bfrag;
      bfrag.x = srow[cbase + laneB0];             // coalesced: 16 consecutive floats/half-wave
      bfrag.y = srow[cbase + laneB1];
      const int k0    = kh * 64 + kc * 4;
      const int abase = (k0 >> 1) * 128 + aLane;
#pragma unroll
      for (int t = 0; t < 4; ++t) {
        v2f afrag = *(const v2f*)(wlds + abase + t * 32);  // ds_load_b64
        acc[t] = __builtin_amdgcn_wmma_f32_16x16x4_f32(
            /*neg_a=*/false, afrag, /*neg_b=*/false, bfrag,
            /*c_mod=*/(short)0, acc[t], /*reuse_a=*/false, /*reuse_b=*/false);
      }
    }
  }
}

__device__ __forceinline__ void decode_tile(int& b, int& h, int& i0, int& n, int& hi) {
  const int lane = threadIdx.x & 31;
  n  = lane & 15;
  hi = lane >> 4;
  const int tile = blockIdx.x * 8 + (threadIdx.x >> 5);   // 8 waves per WG
  const int it = tile & 3;                                // i-tile: {0,16,32,44}
  h  = (tile >> 2) & 511;
  b  = tile >> 11;
  i0 = (it == 3) ? 44 : it * 16;                          // last tile overlaps (identical values)
}

// ---------------------------------------------------------------------------
// Pass 1: down[b,co,h,i] = relu(conv_h(x_i) + bias) + (i<59 ? x[...,i+1] : 0)
// ---------------------------------------------------------------------------
__global__ __launch_bounds__(256) void pass1_kernel(const float* __restrict__ x,
                                                    const float* __restrict__ wt,
                                                    const float* __restrict__ bias,
                                                    float* __restrict__ down) {
  __shared__ float wlds[KTOT * 64];
  load_weights_lds(wt, wlds);

  int b, h, i0, n, hi;
  decode_tile(b, h, i0, n, hi);

  v8f acc[4] = {};
  conv_tile(x, wlds, b, h, i0, n, hi, acc);

  const int  i     = i0 + n;
  const bool sh_ok = (i + 1) < W_;
#pragma unroll
  for (int t = 0; t < 4; ++t) {
#pragma unroll
    for (int v = 0; v < 8; ++v) {
      const int  co  = t * 16 + v + 8 * hi;
      const long off = (((long)b * C_ + co) * H_ + h) * W_ + i;
      float r  = fmaxf(acc[t][v] + bias[co], 0.0f);
      float xs = sh_ok ? x[off + 1] : 0.0f;
      down[off] = r + xs;                                  // contiguous per half-wave
    }
  }
}

// ---------------------------------------------------------------------------
// Pass 2: computes cd[i] = relu(conv_h(down_i)+bias) at columns i = i0..i0+15,
// writes out[k] = down[k] + cd[k+1] at k = i-1 (k >= 0). Column 59 handled by
// col59_kernel. Overlapping tiles write identical values (deterministic).
// ---------------------------------------------------------------------------
__global__ __launch_bounds__(256) void pass2_kernel(const float* __restrict__ down,
                                                    const float* __restrict__ wt,
                                                    const float* __restrict__ bias,
                                                    float* __restrict__ out) {
  __shared__ float wlds[KTOT * 64];
  load_weights_lds(wt, wlds);

  int b, h, i0, n, hi;
  decode_tile(b, h, i0, n, hi);

  v8f acc[4] = {};
  conv_tile(down, wlds, b, h, i0, n, hi, acc);

  const int  k  = i0 + n - 1;
  const bool ok = (k >= 0);
#pragma unroll
  for (int t = 0; t < 4; ++t) {
#pragma unroll
    for (int v = 0; v < 8; ++v) {
      const int  co  = t * 16 + v + 8 * hi;
      const long off = (((long)b * C_ + co) * H_ + h) * W_ + k;
      float cd = fmaxf(acc[t][v] + bias[co], 0.0f);
      if (ok) out[off] = down[off] + cd;
    }
  }
}

// out[..., 59] = down[..., 59]  (no cd term exists for k=59)
__global__ __launch_bounds__(256) void col59_kernel(const float* __restrict__ down,
                                                    float* __restrict__ out) {
  const long idx   = (long)blockIdx.x * 256 + threadIdx.x;   // over B*C*H rows
  const long total = (long)B_ * C_ * H_;
  if (idx < total) {
    const long off = idx * W_ + (W_ - 1);
    out[off] = down[off];
  }
}

extern "C" void kernel_launch(void* const* d_in, const int* in_sizes, int n_in,
                              void* d_out, int out_size, void* d_ws, size_t ws_size,
                              hipStream_t stream) {
  const float* x    = (const float*)d_in[0];   // [16,64,512,60] f32
  const float* w    = (const float*)d_in[1];   // [64,64,3,3] f32
  const float* bias = (const float*)d_in[2];   // [64] f32
  float*       out  = (float*)d_out;           // [16,64,512,60] f32
  float*       down = (float*)d_ws;            // scratch: 16*64*512*60 f32 = 120 MiB

  const int tiles = B_ * H_ * 4;               // 32768 wave-tiles
  const int wgs   = tiles / 8;                 // 8 waves (256 threads) per WG
  dim3 block(256);

  pass1_kernel<<<wgs, block, 0, stream>>>(x, w, bias, down);
  pass2_kernel<<<wgs, block, 0, stream>>>(down, w, bias, out);

  const long rows = (long)B_ * C_ * H_;
  col59_kernel<<<(int)((rows + 255) / 256), block, 0, stream>>>(down, out);
}